// PySigMMDDiscriminator_55250459295936
// MI455X (gfx1250) — compile-verified
//
#include <hip/hip_runtime.h>

typedef __attribute__((ext_vector_type(2))) float v2f;
typedef __attribute__((ext_vector_type(8))) float v8f;

#define LPTS 64        // path length L
#define DDIM 4         // feature dim d (== WMMA K!)
#define NB   32        // batch
#define MM   126       // (L-1)*2^DYADIC_ORDER
#define GSTR 65        // LDS stride for 64x64 Gram
#define WSTR 64        // LDS stride for 63x63 increments

// One workgroup per (combo, a, b) pair: builds the 64x64 RBF Gram with
// v_wmma_f32_16x16x4_f32 (K=4 == d), forms scaled double-increments, then
// runs the 251-step Goursat antidiagonal wavefront entirely in LDS.
__global__ __launch_bounds__(128) void sigker_gram_pde_kernel(
    const float* __restrict__ x, const float* __restrict__ y,
    float* __restrict__ ws) {
  __shared__ float sxx[LPTS];
  __shared__ float syy[LPTS];
  __shared__ float Gs[LPTS * GSTR];     // 64x64 Gram (stride 65)
  __shared__ float Wv[63 * WSTR];       // 0.25*inc_coarse - 1
  __shared__ float diag[3][128];        // rotating wavefront diagonals

  const int tid  = threadIdx.x;
  const int lane = tid & 31;
  const int wave = tid >> 5;

  const int bid   = blockIdx.x;         // 0..3071
  const int combo = bid >> 10;          // 0=xx, 1=xy, 2=yy
  const int pq    = bid & 1023;
  const int a     = pq >> 5;
  const int b     = pq & 31;

  const float* Xa = (combo == 2 ? y : x) + a * (LPTS * DDIM);
  const float* Yb = (combo == 0 ? x : y) + b * (LPTS * DDIM);

  __builtin_prefetch(Xa, 0, 1);         // global_prefetch_b8
  __builtin_prefetch(Yb, 0, 1);

  // ---- squared norms of every path point ----
  if (tid < LPTS) {
    float s = 0.f;
#pragma unroll
    for (int k = 0; k < DDIM; ++k) { float v = Xa[tid * DDIM + k]; s += v * v; }
    sxx[tid] = s;
  } else {
    int t = tid - LPTS;
    float s = 0.f;
#pragma unroll
    for (int k = 0; k < DDIM; ++k) { float v = Yb[t * DDIM + k]; s += v * v; }
    syy[t] = s;
  }

  // ---- A fragment: wave w owns Gram tile-row mi = w ----
  // ISA 16x4 f32 A layout: lanes 0-15 hold K=0..1, lanes 16-31 hold K=2..3.
  const int mi = wave;
  const int rA = lane & 15;
  const int kk = (lane >> 4) << 1;
  v2f afrag;
  afrag.x = Xa[(mi * 16 + rA) * DDIM + kk];
  afrag.y = Xa[(mi * 16 + rA) * DDIM + kk + 1];

  __syncthreads();  // sxx/syy ready

  // ---- 64x64 xy via 4 WMMAs per wave, fused with RBF exp into Gs ----
  const int rowHalf = (lane >> 4) * 8;  // C/D layout: lanes 16-31 hold M=r+8
#pragma unroll
  for (int nj = 0; nj < 4; ++nj) {
    v2f bfrag;   // B (4x16) mirrors A's per-lane pattern for Y rows
    bfrag.x = Yb[(nj * 16 + rA) * DDIM + kk];
    bfrag.y = Yb[(nj * 16 + rA) * DDIM + kk + 1];
    v8f c = {};
    c = __builtin_amdgcn_wmma_f32_16x16x4_f32(
        /*neg_a=*/false, afrag, /*neg_b=*/false, bfrag,
        /*c_mod=*/(short)0, c, /*reuse_a=*/false, /*reuse_b=*/false);
#pragma unroll
    for (int r = 0; r < 8; ++r) {
      int R = mi * 16 + rowHalf + r;
      int C = nj * 16 + rA;
      float d2 = sxx[R] + syy[C] - 2.0f * c[r];
      Gs[R * GSTR + C] = expf(-0.5f * d2);  // sigma = 1
    }
  }
  __syncthreads();

  // ---- coarse double increments, pre-scaled: Wv = 0.25*inc - 1 ----
  for (int idx = tid; idx < 63 * 63; idx += 128) {
    int r = idx / 63, c = idx - r * 63;
    float v = Gs[(r + 1) * GSTR + (c + 1)] - Gs[(r + 1) * GSTR + c]
            - Gs[r * GSTR + (c + 1)]       + Gs[r * GSTR + c];
    Wv[r * WSTR + c] = 0.25f * v - 1.0f;
  }

  // ---- wavefront init: diagonals d=0 and d=1 are boundary ones ----
  diag[0][tid] = (tid == 0) ? 1.0f : 0.0f;
  diag[1][tid] = (tid <= 1) ? 1.0f : 0.0f;
  __syncthreads();

  // ---- Goursat PDE: K[i,j] = K[i-1,j] + K[i,j-1] + K[i-1,j-1]*(inc-1) ----
  const int i   = tid;
  const int im1 = (i == 0) ? 0 : i - 1;
  int curIdx = 2;
  for (int s = 0; s < 2 * MM - 1; ++s) {
    const int d = s + 2;
    const float* p2 = diag[s % 3];
    const float* p1 = diag[(s + 1) % 3];
    curIdx = (s + 2) % 3;
    // inc_fine index (i-1, d-1-i) -> coarse (>>1); clamped reads are masked below
    int r2 = i - 1;     r2 = r2 < 0 ? 0 : (r2 > MM - 1 ? MM - 1 : r2);
    int c2 = d - 1 - i; c2 = c2 < 0 ? 0 : (c2 > MM - 1 ? MM - 1 : c2);
    float wv = Wv[(r2 >> 1) * WSTR + (c2 >> 1)];
    float v = p1[i] + p1[im1] + p2[im1] * wv;
    if (i == 0 || i == d) v = (d <= MM) ? 1.0f : 0.0f;   // boundary cells
    if (i > d || i + MM < d || i > MM) v = 0.0f;         // outside diagonal
    diag[curIdx][i] = v;
    __syncthreads();
  }

  if (tid == 0) ws[bid] = diag[curIdx][MM];  // K[M,M] for this pair
}

// Deterministic combine: mean(kxx) - 2*mean(kxy) + mean(kyy)
__global__ __launch_bounds__(256) void sigker_reduce_kernel(
    const float* __restrict__ ws, float* __restrict__ out) {
  __shared__ float smem[256];
  const int tid = threadIdx.x;
  float acc = 0.f;
  for (int idx = tid; idx < 3 * NB * NB; idx += 256) {
    float w = (idx < NB * NB) ? 1.0f : ((idx < 2 * NB * NB) ? -2.0f : 1.0f);
    acc += w * ws[idx];
  }
  smem[tid] = acc;
  __syncthreads();
  for (int off = 128; off > 0; off >>= 1) {
    if (tid < off) smem[tid] += smem[tid + off];
    __syncthreads();
  }
  if (tid == 0) out[0] = smem[0] * (1.0f / (NB * NB));
}

extern "C" void kernel_launch(void* const* d_in, const int* in_sizes, int n_in,
                              void* d_out, int out_size, void* d_ws, size_t ws_size,
                              hipStream_t stream) {
  (void)in_sizes; (void)n_in; (void)out_size; (void)ws_size;
  const float* x = (const float*)d_in[0];
  const float* y = (const float*)d_in[1];
  float* ws = (float*)d_ws;  // 3*32*32 floats = 12 KB
  sigker_gram_pde_kernel<<<3 * NB * NB, 128, 0, stream>>>(x, y, ws);
  sigker_reduce_kernel<<<1, 256, 0, stream>>>(ws, (float*)d_out);
}